// NormalizingFlow_14439680049712
// MI455X (gfx1250) — compile-verified
//
#include <hip/hip_runtime.h>
#include <hip/hip_bf16.h>
#include <math.h>

typedef float v2f __attribute__((ext_vector_type(2)));
typedef float v4f __attribute__((ext_vector_type(4)));
typedef float v8f __attribute__((ext_vector_type(8)));

#define D 64
#define F 32
#define WAVES 4   // 4 waves * 16 rows = 64 rows per block

// scratch layout (floats): S_masked[1024] | c[32] | wB_frag[2048] | uB_frag[2048]
#define WS_S   0
#define WS_C   1024
#define WS_WB  1056
#define WS_UB  3104

// ---------------------------------------------------------------------------
// Setup: u_hat, strictly-upper-masked S[j][f] = u_hat_j . w_f, c_f = S[f][f],
// and W^T / U_hat pre-swizzled into exact WMMA B-fragment lane order:
//   frag value for (kk, ntile, lane): k = kk*4 + 2*(lane>>4), n = ntile*16 + (lane&15)
//   float2 = { M[k][n], M[k+1][n] }
// ---------------------------------------------------------------------------
__global__ __launch_bounds__(1024) void planar_setup(const float* __restrict__ us,
                                                     const float* __restrict__ ws,
                                                     float* __restrict__ scr) {
  __shared__ float s_uhat[F * D];
  __shared__ float s_coef[F];
  const int tid = threadIdx.x;
  if (tid < F) {
    const float* u = us + tid * D;
    const float* w = ws + tid * D;
    float wu = 0.f, nn = 0.f;
#pragma unroll
    for (int d = 0; d < D; ++d) { wu += u[d] * w[d]; nn += w[d] * w[d]; }
    const float sp = log1pf(expf(wu));            // softplus(wu)
    s_coef[tid] = (sp - 1.f - wu) * rsqrtf(nn);   // scale applied to w/||w||
  }
  __syncthreads();
  for (int idx = tid; idx < F * D; idx += 1024) {
    s_uhat[idx] = us[idx] + s_coef[idx >> 6] * ws[idx];
  }
  __syncthreads();

  // one dot product per thread: S[j][f], masked to the strictly-upper triangle
  {
    const int j = tid >> 5, f = tid & 31;
    const float* uh = s_uhat + j * D;
    const float* w  = ws + f * D;
    float s = 0.f;
#pragma unroll
    for (int d = 0; d < D; ++d) s += uh[d] * w[d];
    scr[WS_S + j * F + f] = (f > j) ? s : 0.f;    // data-masked triangular S
    if (f == j) scr[WS_C + f] = s;                // c_f = w_f . u_hat_f
  }

  // W^T fragments for GEMM1: 16 ksteps x 2 ntiles x 32 lanes (1024 frags)
  {
    const int kk = tid >> 6, rem = tid & 63;
    const int ntile = rem >> 5, lane = rem & 31;
    const int k = kk * 4 + 2 * (lane >> 4);
    const int n = ntile * 16 + (lane & 15);
    scr[WS_WB + tid * 2 + 0] = ws[n * D + k];       // W^T[k][n]
    scr[WS_WB + tid * 2 + 1] = ws[n * D + k + 1];
  }
  // U_hat fragments for GEMM2: 8 ksteps x 4 ntiles x 32 lanes (1024 frags)
  {
    const int kk = tid >> 7, rem = tid & 127;
    const int ntile = rem >> 5, lane = rem & 31;
    const int k = kk * 4 + 2 * (lane >> 4);
    const int n = ntile * 16 + (lane & 15);
    scr[WS_UB + tid * 2 + 0] = s_uhat[k * D + n];   // U_hat[k][n]
    scr[WS_UB + tid * 2 + 1] = s_uhat[(k + 1) * D + n];
  }
}

// ---------------------------------------------------------------------------
// Fused main kernel: per wave, one 16-row tile.
//   GEMM1 (WMMA f32 16x16x4): lin = X*W^T + b
//   32-step triangular recurrence (owner computes h_f, LDS b128 broadcast)
//   GEMM2 (WMMA f32 16x16x4): Z = X + H*U_hat
// ---------------------------------------------------------------------------
__global__ __launch_bounds__(WAVES * 32) void planar_main(
    const float* __restrict__ x, const float* __restrict__ bs,
    const float* __restrict__ scr, float* __restrict__ zout,
    float* __restrict__ ldout) {
  __shared__ float s_frag[4096];          // wB (2048) | uB (2048), lane-major fragments
  __shared__ float s_S [F * F];           // masked S[j][f]
  __shared__ float s_b [F];
  __shared__ float s_c [F];
  __shared__ float s_x [WAVES][16 * 68];  // per-wave x tile / z staging (272B rows, 16B aligned)
  __shared__ float s_hb[WAVES][F * 16];   // per-wave h broadcast buffer [f][row]

  const int tid  = threadIdx.x;
  const int wave = tid >> 5, lane = tid & 31;
  const int lo = lane & 15, hi = lane >> 4;

  const size_t tile = (size_t)blockIdx.x * WAVES + wave;
  const size_t r0   = tile * 16;
  const float* xt   = x + r0 * D;

  __builtin_prefetch(xt + lane * 32, 0, 1);   // global_prefetch_b8 over the 4KB tile

  // ---- stage constants: straight coalesced v4f copies (pre-swizzled in setup) ----
  {
    v4f* df = (v4f*)s_frag;
    const v4f* sf = (const v4f*)(scr + WS_WB);
    for (int i = tid; i < 1024; i += WAVES * 32) df[i] = sf[i];
    v4f* dS = (v4f*)s_S;
    const v4f* sS = (const v4f*)(scr + WS_S);
    for (int i = tid; i < 256; i += WAVES * 32) dS[i] = sS[i];
    if (tid < F) { s_b[tid] = bs[tid]; s_c[tid] = scr[WS_C + tid]; }
  }
  __syncthreads();

  float* sx  = s_x[wave];
  float* shb = s_hb[wave];

  // ---- coalesced tile load: 16 rows x 64 f32 via b128 ----
#pragma unroll
  for (int i = 0; i < 8; ++i) {
    const int r = 2 * i + hi, c = lo * 4;
    *(v4f*)&sx[r * 68 + c] = *(const v4f*)(xt + r * D + c);
  }

  // ---- GEMM1: acc = X*W^T + b,  two N-tiles of 16 flows, K=64 in 16 steps ----
  v8f acc0, acc1;
  {
    const float b0 = s_b[lo], b1 = s_b[16 + lo];
#pragma unroll
    for (int j = 0; j < 8; ++j) { acc0[j] = b0; acc1[j] = b1; }
  }
#pragma unroll
  for (int kk = 0; kk < 16; ++kk) {
    const int kb = kk * 4 + 2 * hi;
    v2f a  = *(const v2f*)&sx[lo * 68 + kb];                          // A: row=lo
    v2f b0 = *(const v2f*)&s_frag[((kk * 2 + 0) * 32 + lane) * 2];    // one b64 each
    v2f b1 = *(const v2f*)&s_frag[((kk * 2 + 1) * 32 + lane) * 2];
    acc0 = __builtin_amdgcn_wmma_f32_16x16x4_f32(false, a, false, b0, (short)0, acc0, false, false);
    acc1 = __builtin_amdgcn_wmma_f32_16x16x4_f32(false, a, false, b1, (short)0, acc1, false, false);
  }

  // ---- sequential triangular recurrence over 32 flows ----
  // acc0[j] = lin[row j+8*hi][flow lo]; acc1 -> flow 16+lo
  float ldp[8];
#pragma unroll
  for (int j = 0; j < 8; ++j) ldp[j] = 0.f;

  for (int f = 0; f < 16; ++f) {          // flows 0..15 live in acc0
    if (lo == f) {
      const float cf = s_c[f];
      v4f hv0, hv1;
#pragma unroll
      for (int j = 0; j < 8; ++j) {
        const float lc = fminf(fmaxf(acc0[j], -15.f), 15.f);
        const float e  = __expf(2.f * lc);
        const float h  = (e - 1.f) * __builtin_amdgcn_rcpf(e + 1.f);  // tanh
        if (j < 4) hv0[j] = h; else hv1[j - 4] = h;
        ldp[j] += __logf(fabsf(1.f + (1.f - h * h) * cf));
      }
      *(v4f*)&shb[f * 16 + 8 * hi]     = hv0;
      *(v4f*)&shb[f * 16 + 8 * hi + 4] = hv1;
    }
    const v4f hb0 = *(const v4f*)&shb[f * 16 + 8 * hi];       // same-wave DS order
    const v4f hb1 = *(const v4f*)&shb[f * 16 + 8 * hi + 4];
    const float sc0 = s_S[f * F + lo];          // masked: zero unless g > f
    const float sc1 = s_S[f * F + 16 + lo];
#pragma unroll
    for (int j = 0; j < 4; ++j) {
      acc0[j]     += sc0 * hb0[j];  acc0[j + 4] += sc0 * hb1[j];
      acc1[j]     += sc1 * hb0[j];  acc1[j + 4] += sc1 * hb1[j];
    }
  }
  for (int f = 16; f < 32; ++f) {         // flows 16..31 live in acc1; acc0 done
    if (lo == (f & 15)) {
      const float cf = s_c[f];
      v4f hv0, hv1;
#pragma unroll
      for (int j = 0; j < 8; ++j) {
        const float lc = fminf(fmaxf(acc1[j], -15.f), 15.f);
        const float e  = __expf(2.f * lc);
        const float h  = (e - 1.f) * __builtin_amdgcn_rcpf(e + 1.f);
        if (j < 4) hv0[j] = h; else hv1[j - 4] = h;
        ldp[j] += __logf(fabsf(1.f + (1.f - h * h) * cf));
      }
      *(v4f*)&shb[f * 16 + 8 * hi]     = hv0;
      *(v4f*)&shb[f * 16 + 8 * hi + 4] = hv1;
    }
    const v4f hb0 = *(const v4f*)&shb[f * 16 + 8 * hi];
    const v4f hb1 = *(const v4f*)&shb[f * 16 + 8 * hi + 4];
    const float sc1 = s_S[f * F + 16 + lo];
#pragma unroll
    for (int j = 0; j < 4; ++j) {
      acc1[j] += sc1 * hb0[j];  acc1[j + 4] += sc1 * hb1[j];
    }
  }

  // ---- GEMM2: Z = X + H*U_hat, four N-tiles of 16 dims, K=32 in 8 steps ----
  v8f z0, z1, z2, z3;
#pragma unroll
  for (int j = 0; j < 8; ++j) {           // C init = X tile (C/D layout)
    const int m = (j + 8 * hi) * 68 + lo;
    z0[j] = sx[m]; z1[j] = sx[m + 16]; z2[j] = sx[m + 32]; z3[j] = sx[m + 48];
  }
#pragma unroll
  for (int kk = 0; kk < 8; ++kk) {
    const int kb = kk * 4 + 2 * hi;
    v2f a;                                 // A = H fragment from broadcast buffer
    a.x = shb[(kb + 0) * 16 + lo];
    a.y = shb[(kb + 1) * 16 + lo];
    v2f b0 = *(const v2f*)&s_frag[2048 + ((kk * 4 + 0) * 32 + lane) * 2];
    v2f b1 = *(const v2f*)&s_frag[2048 + ((kk * 4 + 1) * 32 + lane) * 2];
    v2f b2 = *(const v2f*)&s_frag[2048 + ((kk * 4 + 2) * 32 + lane) * 2];
    v2f b3 = *(const v2f*)&s_frag[2048 + ((kk * 4 + 3) * 32 + lane) * 2];
    z0 = __builtin_amdgcn_wmma_f32_16x16x4_f32(false, a, false, b0, (short)0, z0, false, false);
    z1 = __builtin_amdgcn_wmma_f32_16x16x4_f32(false, a, false, b1, (short)0, z1, false, false);
    z2 = __builtin_amdgcn_wmma_f32_16x16x4_f32(false, a, false, b2, (short)0, z2, false, false);
    z3 = __builtin_amdgcn_wmma_f32_16x16x4_f32(false, a, false, b3, (short)0, z3, false, false);
  }

  // ---- stage Z in LDS, then coalesced b128 stores ----
#pragma unroll
  for (int j = 0; j < 8; ++j) {
    const int m = (j + 8 * hi) * 68 + lo;
    sx[m] = z0[j]; sx[m + 16] = z1[j]; sx[m + 32] = z2[j]; sx[m + 48] = z3[j];
  }
#pragma unroll
  for (int i = 0; i < 8; ++i) {
    const int r = 2 * i + hi, c = lo * 4;
    *(v4f*)(zout + (r0 + r) * D + c) = *(const v4f*)&sx[r * 68 + c];
  }

  // ---- reduce log-det partials across each 16-lane half, store ----
#pragma unroll
  for (int m = 1; m < 16; m <<= 1) {
#pragma unroll
    for (int j = 0; j < 8; ++j) ldp[j] += __shfl_xor(ldp[j], m, 32);
  }
  if (lo == 0) {
#pragma unroll
    for (int j = 0; j < 8; ++j) ldout[r0 + j + 8 * hi] = ldp[j];
  }
}

// ---------------------------------------------------------------------------
extern "C" void kernel_launch(void* const* d_in, const int* in_sizes, int n_in,
                              void* d_out, int out_size, void* d_ws, size_t ws_size,
                              hipStream_t stream) {
  (void)n_in; (void)out_size; (void)ws_size;
  const float* x  = (const float*)d_in[0];
  const float* us = (const float*)d_in[1];
  const float* ws = (const float*)d_in[2];
  const float* bs = (const float*)d_in[3];
  const int B = in_sizes[0] / D;                 // 524288

  float* scr   = (float*)d_ws;
  float* zout  = (float*)d_out;
  float* ldout = zout + (size_t)B * D;

  planar_setup<<<1, 1024, 0, stream>>>(us, ws, scr);
  planar_main<<<B / (WAVES * 16), WAVES * 32, 0, stream>>>(x, bs, scr, zout, ldout);
}